// CDGR_90494960926993
// MI455X (gfx1250) — compile-verified
//
#include <hip/hip_runtime.h>
#include <hip/hip_bf16.h>

#define DEV __device__ __forceinline__

#define BB    16
#define C     256
#define H     32
#define WW    32
#define HW    1024
#define MPHI  16
#define NNODE 20
#define DEMB  300

typedef float v2f __attribute__((ext_vector_type(2)));
typedef float v8f __attribute__((ext_vector_type(8)));

// CDNA5 fp32 WMMA: D(16x16,f32) = A(16x4,f32) * B(4x16,f32) + C
DEV v8f wmma4(v2f a, v2f b, v8f c) {
  return __builtin_amdgcn_wmma_f32_16x16x4_f32(
      false, a, false, b, (short)0, c, false, false);
}

// ---------------------------------------------------------------------------
// Tensor Data Mover (TDM): DMA a 2-D tile (tile_w x tile_h elems of f32) from
// global to LDS with hardware LDS padding (pad_amount DWORDs inserted every
// pad_interval DWORDs -> padded LDS row stride for bank-conflict-free WMMA
// fragment reads). D# layout per CDNA5 ISA ch.8 (CDNA5 ISA).
// ---------------------------------------------------------------------------
#if __has_builtin(__builtin_amdgcn_tensor_load_to_lds)
#define HAVE_TDM 1
#else
#define HAVE_TDM 0
#endif

#if HAVE_TDM
typedef unsigned int u32x4 __attribute__((ext_vector_type(4)));
typedef int i32x8 __attribute__((ext_vector_type(8)));
typedef int i32x4 __attribute__((ext_vector_type(4)));

DEV unsigned lds_byte_off(const void* p) {
  // addrspacecast(LDS->generic) places the LDS byte offset in the low 32 bits
  return (unsigned)(unsigned long long)(uintptr_t)p;
}

DEV void tdm_load_2d_f32(const float* gsrc, unsigned ldsoff,
                         unsigned tile_w, unsigned tile_h,
                         unsigned row_stride_elems,
                         unsigned pad_interval_code,  // 0=2,1=4,...,5=64,7=256 DW
                         unsigned pad_amount_code)    // 0=1DW, 3=4DW
{
  unsigned long long ga = (unsigned long long)(uintptr_t)gsrc;
  u32x4 g0;
  g0[0] = 1u;                                   // count=1 (valid user D#)
  g0[1] = ldsoff;                               // lds_addr (bytes)
  g0[2] = (unsigned)(ga & 0xffffffffu);         // global_addr[31:0]
  g0[3] = (unsigned)((ga >> 32) & 0x01ffffffu)  // global_addr[56:32]
        | (2u << 30);                           // type=2 ("image")
  unsigned td0 = row_stride_elems;              // tensor_dim0 >= tile_w
  unsigned td1 = 0x10000u;                      // tensor_dim1 (tile in-bounds)
  i32x8 g1;
  g1[0] = (int)((2u << 16)                      // data_size=2 -> 4 bytes
              | (1u << 20)                      // pad_enable
              | (pad_interval_code << 22)
              | (pad_amount_code << 25));
  g1[1] = (int)((td0 & 0xffffu) << 16);                           // dim0 lo16
  g1[2] = (int)(((td0 >> 16) & 0xffffu) | ((td1 & 0xffffu) << 16)); // dim0 hi / dim1 lo
  g1[3] = (int)(((td1 >> 16) & 0xffffu) | ((tile_w & 0xffffu) << 16)); // dim1 hi / tile_dim0
  g1[4] = (int)(tile_h & 0xffffu);              // tile_dim1 (tile_dim2=0)
  g1[5] = (int)row_stride_elems;                // tensor_dim0_stride lo32
  g1[6] = 0;                                    // stride hi16 | dim1_stride lo16
  g1[7] = 0;
  i32x4 z4 = {0, 0, 0, 0};
#if defined(__clang_major__) && __clang_major__ >= 23
  i32x8 z8 = {0, 0, 0, 0, 0, 0, 0, 0};
  __builtin_amdgcn_tensor_load_to_lds(g0, g1, z4, z4, z8, 0);
#else
  __builtin_amdgcn_tensor_load_to_lds(g0, g1, z4, z4, 0);
#endif
}
#endif // HAVE_TDM

// ---------------------------------------------------------------------------
// K0: attention over emb (20x300). att@v mean collapses to a weighted emb sum.
// ---------------------------------------------------------------------------
__global__ __launch_bounds__(256) void k0_attention(
    const float* __restrict__ emb, const float* __restrict__ wq, const float* __restrict__ bq,
    const float* __restrict__ wk, const float* __restrict__ bk,
    const float* __restrict__ wv, const float* __restrict__ bv,
    const float* __restrict__ wo, const float* __restrict__ bo,
    float* __restrict__ ev)
{
  __shared__ float q[NNODE*DEMB];
  __shared__ float k[NNODE*DEMB];
  __shared__ float att[NNODE*NNODE];
  __shared__ float wmean[NNODE];
  __shared__ float e2[DEMB];
  __shared__ float n1[DEMB];
  __shared__ float n2[DEMB];
  int tid = threadIdx.x;
  for (int o = tid; o < NNODE*DEMB; o += 256) {
    int n = o / DEMB, d = o % DEMB;
    float aq = bq[d], ak = bk[d];
    for (int kk = 0; kk < DEMB; ++kk) {
      float e = emb[n*DEMB + kk];
      aq += e * wq[kk*DEMB + d];
      ak += e * wk[kk*DEMB + d];
    }
    q[o] = aq; k[o] = ak;
  }
  __syncthreads();
  const float scale = rsqrtf((float)DEMB);
  for (int o = tid; o < NNODE*NNODE; o += 256) {
    int i = o / NNODE, j = o % NNODE;
    float s = 0.f;
    for (int d = 0; d < DEMB; ++d) s += q[i*DEMB+d] * k[j*DEMB+d];
    att[o] = s * scale;
  }
  __syncthreads();
  if (tid < NNODE) {
    float mx = -1e30f;
    for (int j = 0; j < NNODE; ++j) mx = fmaxf(mx, att[tid*NNODE+j]);
    float s = 0.f;
    for (int j = 0; j < NNODE; ++j) { float e = expf(att[tid*NNODE+j]-mx); att[tid*NNODE+j] = e; s += e; }
    float inv = 1.f/s;
    for (int j = 0; j < NNODE; ++j) att[tid*NNODE+j] *= inv;
  }
  __syncthreads();
  if (tid < NNODE) {
    float s = 0.f;
    for (int i = 0; i < NNODE; ++i) s += att[i*NNODE+tid];
    wmean[tid] = s * (1.f/NNODE);
  }
  __syncthreads();
  for (int kk = tid; kk < DEMB; kk += 256) {
    float s = 0.f;
    for (int j = 0; j < NNODE; ++j) s += wmean[j] * emb[j*DEMB+kk];
    e2[kk] = s;
  }
  __syncthreads();
  for (int d = tid; d < DEMB; d += 256) {
    float s = bv[d];
    for (int kk = 0; kk < DEMB; ++kk) s += e2[kk]*wv[kk*DEMB+d];
    n1[d] = s;
  }
  __syncthreads();
  for (int d = tid; d < DEMB; d += 256) {
    float s = bo[d];
    for (int kk = 0; kk < DEMB; ++kk) s += n1[kk]*wo[kk*DEMB+d];
    n2[d] = s;
  }
  __syncthreads();
  for (int o = tid; o < NNODE*DEMB; o += 256)
    ev[o] = emb[o] + n2[o % DEMB];
}

// ---------------------------------------------------------------------------
// K1: adj normalization + g1/g2 (batch-invariant) + derived vvec[o]
// ---------------------------------------------------------------------------
__global__ __launch_bounds__(256) void k1_graph(
    const float* __restrict__ adj, const float* __restrict__ ev,
    const float* __restrict__ gc1, const float* __restrict__ gc2,
    const float* __restrict__ finw,
    float* __restrict__ g2out, float* __restrict__ vvec)
{
  __shared__ float an[NNODE*NNODE];
  __shared__ float dinv[NNODE];
  __shared__ float bufA[NNODE*C];
  __shared__ float bufB[NNODE*C];
  __shared__ float Gr[C];
  int tid = threadIdx.x;
  if (tid < NNODE) {
    float s = 1.f;
    for (int j = 0; j < NNODE; ++j) s += adj[tid*NNODE+j];
    dinv[tid] = rsqrtf(s);
  }
  __syncthreads();
  for (int o = tid; o < NNODE*NNODE; o += 256) {
    int i = o/NNODE, j = o%NNODE;
    an[o] = dinv[i] * (adj[o] + (i==j ? 1.f : 0.f)) * dinv[j];
  }
  __syncthreads();
  for (int o = tid; o < NNODE*C; o += 256) {     // t = ev @ gc1
    int n = o / C, c = o % C;
    float s = 0.f;
    for (int kk = 0; kk < DEMB; ++kk) s += ev[n*DEMB+kk]*gc1[kk*C+c];
    bufA[o] = s;
  }
  __syncthreads();
  for (int o = tid; o < NNODE*C; o += 256) {     // g1 = relu(an @ t)
    int n = o/C, c = o%C;
    float s = 0.f;
    for (int m = 0; m < NNODE; ++m) s += an[n*NNODE+m]*bufA[m*C+c];
    bufB[o] = fmaxf(s, 0.f);
  }
  __syncthreads();
  for (int o = tid; o < NNODE*C; o += 256) {     // u = g1 @ gc2
    int n = o/C, c = o%C;
    float s = 0.f;
    for (int kk = 0; kk < C; ++kk) s += bufB[n*C+kk]*gc2[kk*C+c];
    bufA[o] = s;
  }
  __syncthreads();
  for (int o = tid; o < NNODE*C; o += 256) {     // g2 = relu(an @ u)
    int n = o/C, c = o%C;
    float s = 0.f;
    for (int m = 0; m < NNODE; ++m) s += an[n*NNODE+m]*bufA[m*C+c];
    float g = fmaxf(s, 0.f);
    bufB[o] = g; g2out[o] = g;
  }
  __syncthreads();
  if (tid < C) {                                  // G[c] = relu(sum_n g2[n,c])
    float s = 0.f;
    for (int n = 0; n < NNODE; ++n) s += bufB[n*C+tid];
    Gr[tid] = fmaxf(s, 0.f);
  }
  __syncthreads();
  if (tid < C) {                                  // vvec = final_w[:,256:] @ reluG
    float s = 0.f;
    for (int c = 0; c < C; ++c) s += finw[tid*2*C + C + c]*Gr[c];
    vvec[tid] = s;
  }
}

// ---------------------------------------------------------------------------
// K2: W1 = final_w[:, :256] @ gw_w  (256x256, K=256) via fp32 WMMA
// ---------------------------------------------------------------------------
__global__ __launch_bounds__(32) void k2_fuseW1(
    const float* __restrict__ finw, const float* __restrict__ gww,
    float* __restrict__ W1)
{
  int lane = threadIdx.x & 31;
  int row = lane & 15, col = lane & 15;
  int kb = (lane >> 4) * 2, half = lane >> 4;
  int r0 = blockIdx.y*16, c0 = blockIdx.x*16;
  v8f acc = {};
  for (int kk = 0; kk < C; kk += 4) {
    v2f a, b;
    a.x = finw[(r0+row)*(2*C) + kk + kb];
    a.y = finw[(r0+row)*(2*C) + kk + kb + 1];
    b.x = gww[(kk+kb  )*C + c0 + col];
    b.y = gww[(kk+kb+1)*C + c0 + col];
    acc = wmma4(a, b, acc);
  }
  for (int i = 0; i < 8; ++i)
    W1[(r0 + i + half*8)*C + c0 + col] = acc[i];
}

// ---------------------------------------------------------------------------
// K3: phi[b,m,p] (1x1 conv, 16 out) + pixel logits a[b,p] = x[:,p]·w_in
// ---------------------------------------------------------------------------
__global__ __launch_bounds__(128) void k3_phi(
    const float* __restrict__ x, const float* __restrict__ phiw,
    const float* __restrict__ phib, const float* __restrict__ s2lw,
    float* __restrict__ phi, float* __restrict__ alog)
{
  __shared__ float pw[MPHI*C];
  __shared__ float win[C];
  int tid = threadIdx.x;
  for (int i = tid; i < MPHI*C; i += 128) pw[i] = phiw[i];
  for (int i = tid; i < C; i += 128) win[i] = s2lw[i];
  __syncthreads();
  int b = blockIdx.y;
  int p = blockIdx.x*128 + tid;
  const float* xb = x + (size_t)b*C*HW;
  float pacc[MPHI];
  #pragma unroll
  for (int m = 0; m < MPHI; ++m) pacc[m] = 0.f;
  float aacc = 0.f;
  for (int c = 0; c < C; ++c) {
    float xv = xb[c*HW + p];
    aacc += xv * win[c];
    #pragma unroll
    for (int m = 0; m < MPHI; ++m) pacc[m] += xv * pw[m*C + c];
  }
  float* pb = phi + (size_t)b*MPHI*HW;
  #pragma unroll
  for (int m = 0; m < MPHI; ++m) pb[m*HW + p] = pacc[m] + phib[m];
  alog[b*HW + p] = aacc;
}

// ---------------------------------------------------------------------------
// K3b: channel means -> g -> sigmoid diag values sg[b,m]
// ---------------------------------------------------------------------------
__global__ __launch_bounds__(256) void k3b_glob(
    const float* __restrict__ x, const float* __restrict__ globw,
    float* __restrict__ sg)
{
  __shared__ float mean[C];
  int tid = threadIdx.x, b = blockIdx.x;
  const float* xb = x + (size_t)b*C*HW;
  float s = 0.f;
  for (int p = 0; p < HW; ++p) s += xb[tid*HW + p];
  mean[tid] = s * (1.f/HW);
  __syncthreads();
  if (tid < MPHI) {
    float g = 0.f;
    for (int c = 0; c < C; ++c) g += globw[tid*C + c]*mean[c];
    sg[b*MPHI + tid] = 1.f/(1.f + expf(-g));
  }
}

// ---------------------------------------------------------------------------
// K4: s[b,p] = softmax_p(a[b,p])  (the g2@w_ev + bias terms cancel over p)
// ---------------------------------------------------------------------------
__global__ __launch_bounds__(256) void k4_softmax_s(
    const float* __restrict__ alog, float* __restrict__ s)
{
  __shared__ float red[256];
  int tid = threadIdx.x, b = blockIdx.x;
  const float* ab = alog + b*HW;
  float mx = -1e30f;
  for (int p = tid; p < HW; p += 256) mx = fmaxf(mx, ab[p]);
  red[tid] = mx; __syncthreads();
  for (int st = 128; st > 0; st >>= 1) { if (tid < st) red[tid] = fmaxf(red[tid], red[tid+st]); __syncthreads(); }
  mx = red[0]; __syncthreads();
  float sm = 0.f;
  for (int p = tid; p < HW; p += 256) sm += expf(ab[p]-mx);
  red[tid] = sm; __syncthreads();
  for (int st = 128; st > 0; st >>= 1) { if (tid < st) red[tid] += red[tid+st]; __syncthreads(); }
  float inv = 1.f/red[0];
  for (int p = tid; p < HW; p += 256) s[b*HW+p] = expf(ab[p]-mx)*inv;
}

// ---------------------------------------------------------------------------
// K5: Y[b,p,m] = 0.5*rowsum(x_phi[p,:]) + x_phi[p,m]*(sg[m]-0.5)
// ---------------------------------------------------------------------------
__global__ __launch_bounds__(256) void k5_Y(
    const float* __restrict__ phi, const float* __restrict__ sg,
    float* __restrict__ Y)
{
  int b = blockIdx.y;
  int p = blockIdx.x*256 + threadIdx.x;
  const float* pb = phi + (size_t)b*MPHI*HW + (size_t)p*MPHI;
  float v[MPHI]; float R = 0.f;
  #pragma unroll
  for (int m = 0; m < MPHI; ++m) { float t = fmaxf(pb[m], 0.f); v[m] = t; R += t; }
  float* yb = Y + (size_t)b*HW*MPHI + (size_t)p*MPHI;
  #pragma unroll
  for (int m = 0; m < MPHI; ++m) yb[m] = 0.5f*R + v[m]*(sg[b*MPHI+m]-0.5f);
}

// ---------------------------------------------------------------------------
// K6: A[b] = softmax_rows( Y(1024x16) @ reluPhi(16x1024) ), WMMA K=16.
// ---------------------------------------------------------------------------
__global__ __launch_bounds__(256) void k6_A(
    const float* __restrict__ phi, const float* __restrict__ Y,
    float* __restrict__ Amat)
{
  __shared__ float P[16*1024];
  __shared__ float red[256];
  __shared__ float rowred[16];
  int tid = threadIdx.x;
  int lane = tid & 31, wave = tid >> 5;
  int b = blockIdx.y, r0 = blockIdx.x*16;
  const float* pb = phi + (size_t)b*MPHI*HW;
  const float* yb = Y + (size_t)b*HW*MPHI;
  int rowA = lane & 15, kb = (lane >> 4)*2;
  int colB = lane & 15, half = lane >> 4;
  for (int t8 = 0; t8 < 8; ++t8) {
    int qt = wave*8 + t8;
    v8f acc = {};
    #pragma unroll
    for (int kk = 0; kk < MPHI; kk += 4) {
      v2f a, bb;
      a.x = yb[(r0+rowA)*MPHI + kk + kb];
      a.y = yb[(r0+rowA)*MPHI + kk + kb + 1];
      int qq = qt*16 + colB;
      bb.x = fmaxf(pb[(kk+kb  )*HW + qq], 0.f);
      bb.y = fmaxf(pb[(kk+kb+1)*HW + qq], 0.f);
      acc = wmma4(a, bb, acc);
    }
    #pragma unroll
    for (int i = 0; i < 8; ++i)
      P[(i + half*8)*1024 + qt*16 + colB] = acc[i];
  }
  __syncthreads();
  int r = tid >> 4, l16 = tid & 15;
  float mx = -1e30f;
  for (int j = l16; j < 1024; j += 16) mx = fmaxf(mx, P[r*1024+j]);
  red[tid] = mx; __syncthreads();
  if (l16 == 0) {
    float m2 = red[tid];
    for (int t = 1; t < 16; ++t) m2 = fmaxf(m2, red[tid+t]);
    rowred[r] = m2;
  }
  __syncthreads();
  mx = rowred[r];
  float sm = 0.f;
  for (int j = l16; j < 1024; j += 16) {
    float e = expf(P[r*1024+j] - mx);
    P[r*1024+j] = e; sm += e;
  }
  red[tid] = sm; __syncthreads();
  if (l16 == 0) {
    float s2 = 0.f;
    for (int t = 0; t < 16; ++t) s2 += red[tid+t];
    rowred[r] = s2;
  }
  __syncthreads();
  float inv = 1.f / rowred[r];
  float* Ab = Amat + (size_t)b*HW*HW + (size_t)(r0+r)*HW;
  for (int j = l16; j < 1024; j += 16) Ab[j] = P[r*1024+j]*inv;
}

// ---------------------------------------------------------------------------
// K7: spiral = x_flat - A @ x_flat   (M=1024,N=256,K=1024 per batch)
//     TDM double-buffered A-strip staging (padded stride 65) + fp32 WMMA.
// ---------------------------------------------------------------------------
__global__ __launch_bounds__(256) void k7_spiral(
    const float* __restrict__ x, const float* __restrict__ Amat,
    float* __restrict__ spiral)
{
  __shared__ float At[2][16*65];
  int tid = threadIdx.x, lane = tid & 31, wave = tid >> 5;
  int b = blockIdx.z, r0 = blockIdx.y*16;
  int c0 = blockIdx.x*128 + wave*16;
  const float* xb = x + (size_t)b*C*HW;          // flat view: [q*256 + c]
  const float* Ab = Amat + (size_t)b*HW*HW;
  int rowA = lane & 15, kb = (lane>>4)*2, colB = lane & 15, half = lane>>4;
  v8f acc = {};
#if HAVE_TDM
  // Preload tile kk=0 via the Tensor Data Mover (one wave issues the DMA).
  if (wave == 0) {
    tdm_load_2d_f32(Ab + (size_t)r0*HW, lds_byte_off(&At[0][0]),
                    /*tile_w=*/64, /*tile_h=*/16, /*stride=*/HW,
                    /*pad_interval(64DW)=*/5, /*pad_amount(1DW)=*/0);
    __builtin_amdgcn_s_wait_tensorcnt(0);
  }
  __syncthreads();
  int cur = 0;
  for (int kk = 0; kk < HW; kk += 64) {
    // Kick off DMA of the next A strip while this one feeds the WMMAs.
    if (wave == 0 && kk + 64 < HW)
      tdm_load_2d_f32(Ab + (size_t)r0*HW + kk + 64,
                      lds_byte_off(&At[cur ^ 1][0]), 64, 16, HW, 5, 0);
    #pragma unroll
    for (int k4 = 0; k4 < 64; k4 += 4) {
      v2f a, bb;
      a.x = At[cur][rowA*65 + k4 + kb];
      a.y = At[cur][rowA*65 + k4 + kb + 1];
      int q = kk + k4 + kb;
      bb.x = xb[(size_t)q*C + c0 + colB];
      bb.y = xb[(size_t)(q+1)*C + c0 + colB];
      acc = wmma4(a, bb, acc);
    }
    if (wave == 0) __builtin_amdgcn_s_wait_tensorcnt(0);
    __syncthreads();
    cur ^= 1;
  }
#else
  for (int kk = 0; kk < HW; kk += 64) {
    __syncthreads();
    #pragma unroll
    for (int j = 0; j < 4; ++j) {
      int f = tid + j*256;
      int r = f >> 6, c = f & 63;
      At[0][r*65 + c] = Ab[(size_t)(r0+r)*HW + kk + c];
      if (kk + 64 < HW)
        __builtin_prefetch(&Ab[(size_t)(r0+r)*HW + kk + 64 + c], 0, 0);
    }
    __syncthreads();
    #pragma unroll
    for (int k4 = 0; k4 < 64; k4 += 4) {
      v2f a, bb;
      a.x = At[0][rowA*65 + k4 + kb];
      a.y = At[0][rowA*65 + k4 + kb + 1];
      int q = kk + k4 + kb;
      bb.x = xb[(size_t)q*C + c0 + colB];
      bb.y = xb[(size_t)(q+1)*C + c0 + colB];
      acc = wmma4(a, bb, acc);
    }
  }
#endif
  float* sb = spiral + (size_t)b*HW*C;
  #pragma unroll
  for (int i = 0; i < 8; ++i) {
    int rr = r0 + i + half*8;
    int cc = c0 + colB;
    sb[(size_t)rr*C + cc] = xb[(size_t)rr*C + cc] - acc[i];
  }
}

// ---------------------------------------------------------------------------
// K8: out = relu( W1 @ spiral^T + vvec ⊗ s + x )   (256x1024, K=256, WMMA)
//     W1 strip staged via TDM with hardware padding (stride 260).
// ---------------------------------------------------------------------------
__global__ __launch_bounds__(256) void k8_out(
    const float* __restrict__ x, const float* __restrict__ spiral,
    const float* __restrict__ W1, const float* __restrict__ vvec,
    const float* __restrict__ s, float* __restrict__ out)
{
  __shared__ float Ws[16*260];
  int tid = threadIdx.x, lane = tid & 31, wave = tid >> 5;
  int b = blockIdx.z, o0 = blockIdx.y*16;
  int p0 = blockIdx.x*128 + wave*16;
#if HAVE_TDM
  if (wave == 0) {
    tdm_load_2d_f32(W1 + (size_t)o0*C, lds_byte_off(&Ws[0]),
                    /*tile_w=*/C, /*tile_h=*/16, /*stride=*/C,
                    /*pad_interval(256DW)=*/7, /*pad_amount(4DW)=*/3);
    __builtin_amdgcn_s_wait_tensorcnt(0);
  }
  __syncthreads();
#else
  for (int i = tid; i < 16*C; i += 256) {
    int r = i >> 8, c = i & 255;
    Ws[r*260 + c] = W1[(o0+r)*C + c];
  }
  __syncthreads();
#endif
  const float* sb = spiral + (size_t)b*HW*C;
  int rowA = lane & 15, kb = (lane>>4)*2, colB = lane & 15, half = lane>>4;
  int p = p0 + colB;
  v8f acc = {};
  #pragma unroll 4
  for (int kk = 0; kk < C; kk += 4) {
    v2f a, bb;
    a.x = Ws[rowA*260 + kk + kb];
    a.y = Ws[rowA*260 + kk + kb + 1];
    bb.x = sb[(size_t)p*C + kk + kb];       // B = spiral^T
    bb.y = sb[(size_t)p*C + kk + kb + 1];
    acc = wmma4(a, bb, acc);
  }
  const float* xb = x + (size_t)b*C*HW;
  float* ob = out + (size_t)b*C*HW;
  float sp = s[b*HW + p];
  #pragma unroll
  for (int i = 0; i < 8; ++i) {
    int o = o0 + i + half*8;
    float val = acc[i] + vvec[o]*sp + xb[(size_t)o*HW + p];
    ob[(size_t)o*HW + p] = fmaxf(val, 0.f);
  }
}

// ---------------------------------------------------------------------------
extern "C" void kernel_launch(void* const* d_in, const int* in_sizes, int n_in,
                              void* d_out, int out_size, void* d_ws, size_t ws_size,
                              hipStream_t stream) {
  (void)in_sizes; (void)n_in; (void)out_size; (void)ws_size;
  const float* x     = (const float*)d_in[0];
  const float* emb   = (const float*)d_in[1];
  const float* adj   = (const float*)d_in[2];
  const float* wq    = (const float*)d_in[3];
  const float* bq    = (const float*)d_in[4];
  const float* wk    = (const float*)d_in[5];
  const float* bk    = (const float*)d_in[6];
  const float* wv    = (const float*)d_in[7];
  const float* bv    = (const float*)d_in[8];
  const float* wo    = (const float*)d_in[9];
  const float* bo    = (const float*)d_in[10];
  const float* phiw  = (const float*)d_in[11];
  const float* phib  = (const float*)d_in[12];
  const float* globw = (const float*)d_in[13];
  const float* gc1   = (const float*)d_in[14];
  const float* gc2   = (const float*)d_in[15];
  const float* gww   = (const float*)d_in[16];
  const float* s2lw  = (const float*)d_in[17];
  // d_in[18] (s2l_b) cancels in the pixel-axis softmax
  const float* finw  = (const float*)d_in[19];
  float* out = (float*)d_out;

  float* W   = (float*)d_ws;              // workspace layout (floats)
  float* EV  = W;                         // 20x300
  float* G2  = W + 6016;                  // 20x256
  float* VV  = W + 11136;                 // 256
  float* SG  = W + 11392;                 // B x 16
  float* AL  = W + 11648;                 // B x 1024
  float* SS  = W + 28032;                 // B x 1024
  float* W1  = W + 44416;                 // 256x256
  float* PHI = W + 109952;                // B x 16 x 1024
  float* Yb  = W + 372096;                // B x 1024 x 16
  float* SP  = W + 634240;                // B x 1024 x 256
  float* AM  = W + 4828544;               // B x 1024 x 1024 (fits in 192MB L2)

  k0_attention<<<1, 256, 0, stream>>>(emb, wq, bq, wk, bk, wv, bv, wo, bo, EV);
  k1_graph    <<<1, 256, 0, stream>>>(adj, EV, gc1, gc2, finw, G2, VV);
  k2_fuseW1   <<<dim3(16,16), 32, 0, stream>>>(finw, gww, W1);
  k3_phi      <<<dim3(8,BB), 128, 0, stream>>>(x, phiw, phib, s2lw, PHI, AL);
  k3b_glob    <<<BB, 256, 0, stream>>>(x, globw, SG);
  k4_softmax_s<<<BB, 256, 0, stream>>>(AL, SS);
  k5_Y        <<<dim3(4,BB), 256, 0, stream>>>(PHI, SG, Yb);
  k6_A        <<<dim3(64,BB), 256, 0, stream>>>(PHI, Yb, AM);
  k7_spiral   <<<dim3(2,64,BB), 256, 0, stream>>>(x, AM, SP);
  k8_out      <<<dim3(8,16,BB), 256, 0, stream>>>(x, SP, W1, VV, SS, out);
}